// lstm_encoder_35983236006406
// MI455X (gfx1250) — compile-verified
//
#include <hip/hip_runtime.h>
#include <math.h>

typedef __attribute__((ext_vector_type(2))) float v2f;
typedef __attribute__((ext_vector_type(8))) float v8f;

#define T_STEPS 384
#define NFEAT   1536
#define HDIM    1536
#define GROWS   6144   // 4*H
#define KDIM    1536

// workspace layout in floats
#define OFF_ATTN 0
#define OFF_WIP  1536
#define OFF_GIN  (OFF_WIP + NFEAT * T_STEPS)           // 591360
#define OFF_H    (OFF_GIN + GROWS * T_STEPS)           // 2950656
#define OFF_BAR  (OFF_H + HDIM)                        // barrier int lives here

#define RECUR_BLOCKS 192   // 8 h-elements per block, 32 gate rows per block

// ---------------------------------------------------------------------------
// Kernel 1: base = input_^T @ w_t + b_att ; attn = softmax(base)
// (softmax is shift-invariant, so the per-step scalar h.w_h + c.w_c cancels)
// ---------------------------------------------------------------------------
__global__ void attn_kernel(const float* __restrict__ in,
                            const float* __restrict__ W_att,
                            const float* __restrict__ b_att,
                            float* __restrict__ attn) {
  __shared__ float sb[NFEAT];
  __shared__ float red[256];
  __shared__ float sval;
  const int tid = threadIdx.x;
  const float* wt = W_att + 2 * HDIM;   // w_t part of W_att row
  const float b0 = b_att[0];

  for (int k = tid; k < NFEAT; k += 256) {
    float s = 0.f;
    for (int t = 0; t < T_STEPS; ++t)
      s = fmaf(in[t * NFEAT + k], wt[t], s);
    sb[k] = s + b0;
  }
  __syncthreads();

  // block max
  float m = -INFINITY;
  for (int k = tid; k < NFEAT; k += 256) m = fmaxf(m, sb[k]);
  red[tid] = m;
  __syncthreads();
  for (int s = 128; s > 0; s >>= 1) {
    if (tid < s) red[tid] = fmaxf(red[tid], red[tid + s]);
    __syncthreads();
  }
  if (tid == 0) sval = red[0];
  __syncthreads();
  const float mx = sval;

  // exp + sum
  float acc = 0.f;
  for (int k = tid; k < NFEAT; k += 256) {
    float e = __expf(sb[k] - mx);
    sb[k] = e;
    acc += e;
  }
  red[tid] = acc;
  __syncthreads();
  for (int s = 128; s > 0; s >>= 1) {
    if (tid < s) red[tid] += red[tid + s];
    __syncthreads();
  }
  if (tid == 0) sval = red[0];
  __syncthreads();
  const float inv = 1.f / sval;
  for (int k = tid; k < NFEAT; k += 256) attn[k] = sb[k] * inv;
}

// ---------------------------------------------------------------------------
// Kernel 2: wi_t = attn ⊙ x_t  -> input_weight output (t-major) and a packed
// (K/2, T, 2) float2 layout for the WMMA B-matrix fetches.
// ---------------------------------------------------------------------------
__global__ void wi_kernel(const float* __restrict__ in,
                          const float* __restrict__ attn,
                          float* __restrict__ out_w,
                          float* __restrict__ wip) {
  const int idx = blockIdx.x * 256 + threadIdx.x;   // idx = t*NFEAT + k
  if (idx >= T_STEPS * NFEAT) return;
  const int t = idx / NFEAT;
  const int k = idx - t * NFEAT;
  const float v = attn[k] * in[idx];
  out_w[idx] = v;                                       // input_weight[t][k]
  wip[((k >> 1) * T_STEPS + t) * 2 + (k & 1)] = v;      // packed pairs
}

// ---------------------------------------------------------------------------
// Kernel 3: Gin = W_ih @ WI^T + (b_ih + b_hh)  via V_WMMA_F32_16X16X4_F32.
// Each wave computes a 64(M) x 16(N) tile: 4 accumulators sharing one B fetch.
// A 16x4 tile = v2f/lane (lanes 0-15: K=0,1 ; lanes 16-31: K=2,3).
// B 4x16 tile = v2f/lane from the packed (K/2,T,2) buffer.
// ---------------------------------------------------------------------------
__global__ void gemm_kernel(const float* __restrict__ W_ih,
                            const float* __restrict__ wip,
                            const float* __restrict__ b_ih,
                            const float* __restrict__ b_hh,
                            float* __restrict__ gin) {
  const int wave = (blockIdx.x * blockDim.x + threadIdx.x) >> 5;
  if (wave >= (GROWS / 64) * (T_STEPS / 16)) return;
  const int lane   = threadIdx.x & 31;
  const int mgroup = wave % (GROWS / 64);   // 96 groups of 64 rows
  const int ngroup = wave / (GROWS / 64);   // 24 groups of 16 cols
  const int r0   = mgroup * 64;
  const int t0   = ngroup * 16;
  const int half = lane >> 4;               // 0: K=0,1 ; 1: K=2,3
  const int mrow = lane & 15;

  v8f acc[4] = {};
  const float* ap[4];
#pragma unroll
  for (int s = 0; s < 4; ++s)
    ap[s] = W_ih + (size_t)(r0 + s * 16 + mrow) * KDIM + 2 * half;
  // B element (pair p = k/2 + half, col t) lives at wip + p*2*T + t*2
  const float* bp = wip + (size_t)half * 2 * T_STEPS + (size_t)(t0 + mrow) * 2;

  for (int k = 0; k < KDIM; k += 4) {
    v2f b = *(const v2f*)(bp + (size_t)k * T_STEPS);
#pragma unroll
    for (int s = 0; s < 4; ++s) {
      v2f a = *(const v2f*)(ap[s] + k);
      acc[s] = __builtin_amdgcn_wmma_f32_16x16x4_f32(
          false, a, false, b, (short)0, acc[s], false, false);
    }
  }

  // C/D layout: VGPR v, lanes 0-15 -> M=v, lanes 16-31 -> M=8+v, N=lane&15
#pragma unroll
  for (int s = 0; s < 4; ++s) {
#pragma unroll
    for (int v = 0; v < 8; ++v) {
      const int row = r0 + s * 16 + v + 8 * half;
      gin[(size_t)row * T_STEPS + t0 + mrow] =
          acc[s][v] + b_ih[row] + b_hh[row];
    }
  }
}

// ---------------------------------------------------------------------------
// Kernel 4: persistent LSTM recurrence. 192 blocks; block b owns h-slice
// [8b, 8b+8) and its 32 gate rows. c stays in LDS for the whole scan; h is
// exchanged through a global buffer with fence + atomic grid barrier per step.
// All W_hh / h traffic is B128 (float4) transactions.
// ---------------------------------------------------------------------------
__device__ __forceinline__ float sigf(float x) {
  return 1.f / (1.f + __expf(-x));
}

__global__ void recur_kernel(const float* __restrict__ W_hh,
                             const float* __restrict__ gin,
                             float* __restrict__ hbuf,
                             float* __restrict__ out_encode,
                             int* __restrict__ bar) {
  __shared__ float h_s[HDIM];
  __shared__ float gates_s[32];
  __shared__ float c_s[8];
  const int tid  = threadIdx.x;
  const int lane = tid & 31;
  const int wv   = tid >> 5;
  const int j0   = blockIdx.x * 8;

  if (tid < 8) c_s[tid] = 0.f;

  const float4* __restrict__ hg4 = (const float4*)hbuf;
  float4* h_s4 = (float4*)h_s;

  for (int t = 0; t < T_STEPS; ++t) {
    // pull the (device-visible) h vector into LDS with B128 transactions
    for (int k = tid; k < HDIM / 4; k += 256) h_s4[k] = hg4[k];
    __syncthreads();

    // 32 gate rows for this block; each wave does 4 B128-vectorized dots
#pragma unroll 1
    for (int rr = 0; rr < 4; ++rr) {
      const int r32 = wv * 4 + rr;
      const int g   = r32 >> 3;
      const int jj  = r32 & 7;
      const int row = g * HDIM + j0 + jj;
      const float4* wr4 = (const float4*)(W_hh + (size_t)row * HDIM);
      if (rr < 3) {
        // prefetch a slice of the next row into near caches
        const int nrow = (g + ((jj + 1) >> 3)) * HDIM + j0 + ((jj + 1) & 7);
        __builtin_prefetch(W_hh + (size_t)nrow * HDIM + lane * 32, 0, 0);
      }
      float acc = 0.f;
#pragma unroll
      for (int i = 0; i < HDIM / 128; ++i) {     // 12 iterations
        const int k4 = i * 32 + lane;
        const float4 w = wr4[k4];
        const float4 h = h_s4[k4];
        acc = fmaf(w.x, h.x, acc);
        acc = fmaf(w.y, h.y, acc);
        acc = fmaf(w.z, h.z, acc);
        acc = fmaf(w.w, h.w, acc);
      }
#pragma unroll
      for (int off = 16; off > 0; off >>= 1) acc += __shfl_xor(acc, off, 32);
      if (lane == 0)
        gates_s[r32] = acc + gin[(size_t)row * T_STEPS + t];
    }
    __syncthreads();

    if (tid < 8) {
      const int jj = tid;
      const float iv = sigf(gates_s[jj]);
      const float fv = sigf(gates_s[8 + jj]);
      const float gv = tanhf(gates_s[16 + jj]);
      const float ov = sigf(gates_s[24 + jj]);
      const float c  = fv * c_s[jj] + iv * gv;
      c_s[jj] = c;
      const float hn = ov * tanhf(c);
      hbuf[j0 + jj] = hn;
      out_encode[(size_t)t * HDIM + j0 + jj] = hn;
    }
    __threadfence();          // make h stores device-visible before signaling
    __syncthreads();

    if (tid == 0) {
      __hip_atomic_fetch_add(bar, 1, __ATOMIC_ACQ_REL, __HIP_MEMORY_SCOPE_AGENT);
      const int target = (t + 1) * RECUR_BLOCKS;
      while (__hip_atomic_load(bar, __ATOMIC_ACQUIRE, __HIP_MEMORY_SCOPE_AGENT) < target)
        __builtin_amdgcn_s_sleep(2);
    }
    __syncthreads();
    __threadfence();          // acquire side: discard stale cached h lines
  }
}

// ---------------------------------------------------------------------------
extern "C" void kernel_launch(void* const* d_in, const int* in_sizes, int n_in,
                              void* d_out, int out_size, void* d_ws, size_t ws_size,
                              hipStream_t stream) {
  const float* input_ = (const float*)d_in[0];
  const float* W_att  = (const float*)d_in[1];
  const float* b_att  = (const float*)d_in[2];
  const float* W_ih   = (const float*)d_in[3];
  const float* W_hh   = (const float*)d_in[4];
  const float* b_ih   = (const float*)d_in[5];
  const float* b_hh   = (const float*)d_in[6];
  (void)in_sizes; (void)n_in; (void)out_size; (void)ws_size;

  float* ws   = (float*)d_ws;
  float* attn = ws + OFF_ATTN;
  float* wip  = ws + OFF_WIP;
  float* gin  = ws + OFF_GIN;
  float* hbuf = ws + OFF_H;
  int*   bar  = (int*)(ws + OFF_BAR);
  float* out  = (float*)d_out;

  // zero h0 and the grid-barrier counter (capture-legal stream memset)
  hipMemsetAsync(hbuf, 0, (HDIM + 16) * sizeof(float), stream);

  attn_kernel<<<1, 256, 0, stream>>>(input_, W_att, b_att, attn);
  wi_kernel<<<(T_STEPS * NFEAT) / 256, 256, 0, stream>>>(input_, attn, out, wip);
  gemm_kernel<<<288, 256, 0, stream>>>(W_ih, wip, b_ih, b_hh, gin);
  recur_kernel<<<RECUR_BLOCKS, 256, 0, stream>>>(
      W_hh, gin, hbuf, out + (size_t)T_STEPS * NFEAT, bar);
}